// MobileFacenetUnsupervisedLoss_4303557231438
// MI455X (gfx1250) — compile-verified
//
#include <hip/hip_runtime.h>
#include <hip/hip_bf16.h>
#include <math.h>

typedef __attribute__((ext_vector_type(2))) float v2f;
typedef __attribute__((ext_vector_type(8))) float v8f;

#define NROWS     131072
#define NCLASS    1024
#define NDIM      128
#define NBLOCKS   (NROWS / 16)   // 8192
#define LG_STRIDE 1044           // 1024 + 20 pad: half-waves hit disjoint LDS banks

// ---------------------------------------------------------------------------
// P1: per-class squared norms of m, reciprocal scales
// ---------------------------------------------------------------------------
__global__ void precompute_kernel(const float* __restrict__ m,
                                  const float* __restrict__ s,
                                  float* __restrict__ m2,
                                  float* __restrict__ invs) {
    int k = blockIdx.x * blockDim.x + threadIdx.x;
    if (k < NCLASS) {
        const float4* mp = (const float4*)(m + (size_t)k * NDIM);
        float acc = 0.0f;
        #pragma unroll 8
        for (int i = 0; i < NDIM / 4; ++i) {
            float4 a = mp[i];
            acc += a.x * a.x + a.y * a.y + a.z * a.z + a.w * a.w;
        }
        m2[k]   = acc;
        invs[k] = 1.0f / s[k];
    }
}

// ---------------------------------------------------------------------------
// P2: inter term -> workspace (combined with entropy partials at the end)
// ---------------------------------------------------------------------------
__global__ void inter_kernel(const float* __restrict__ m,
                             float* __restrict__ interv) {
    __shared__ float red[NDIM];
    int d = threadIdx.x;                       // 128 threads, one per feature dim
    float sum = 0.0f;
    for (int k = 0; k < NCLASS; ++k) sum += m[(size_t)k * NDIM + d];
    float mean = sum * (1.0f / (float)NCLASS);
    float ssq = 0.0f;
    for (int k = 0; k < NCLASS; ++k) {
        float v = m[(size_t)k * NDIM + d] - mean;
        ssq += v * v;
    }
    red[d] = ssq;
    __syncthreads();
    if (d == 0) {
        float t = 0.0f;
        for (int i = 0; i < NDIM; ++i) t += red[i];
        interv[0] = t * (1.0f / (float)NCLASS);
    }
}

// ---------------------------------------------------------------------------
// Main fused kernel: 16 rows per workgroup, 8 waves x 128 classes each.
// dot(x,m) via V_WMMA_F32_16X16X4_F32, then dist/softmax/entropy in LDS.
// One partial (sum of 16 row entropies) per block -> no same-address atomics.
// ---------------------------------------------------------------------------
__global__ __launch_bounds__(256)
void entropy_wmma_kernel(const float* __restrict__ x,
                         const float* __restrict__ m,
                         const float* __restrict__ m2,
                         const float* __restrict__ invs,
                         float* __restrict__ partial) {
    extern __shared__ float smem[];
    float* xtile = smem;                 // 16 x 128 floats (reused as reduce scratch)
    float* x2row = smem + 2048;          // 16 floats
    float* lg    = smem + 2064;          // 16 x LG_STRIDE floats

    const int tid   = threadIdx.x;
    const int wave  = tid >> 5;          // 0..7
    const int lane  = tid & 31;
    const int half  = lane >> 4;         // 0 / 1
    const int l16   = lane & 15;
    const int rbase = blockIdx.x * 16;

    if (tid < 16) x2row[tid] = 0.0f;
    __syncthreads();

    // ---- stage x tile in LDS + accumulate row norms (ds_add_f32) ----
    {
        const float4* xg = (const float4*)(x + (size_t)rbase * NDIM);
        float4 a0 = xg[tid];             // flat float offset 4*tid      -> row tid/32
        float4 a1 = xg[tid + 256];       // flat float offset 4*tid+1024 -> row tid/32+8
        ((float4*)xtile)[tid]       = a0;
        ((float4*)xtile)[tid + 256] = a1;
        float s0 = a0.x * a0.x + a0.y * a0.y + a0.z * a0.z + a0.w * a0.w;
        float s1 = a1.x * a1.x + a1.y * a1.y + a1.z * a1.z + a1.w * a1.w;
        atomicAdd(&x2row[(tid >> 5)],     s0);
        atomicAdd(&x2row[(tid >> 5) + 8], s1);
    }
    __syncthreads();

    // ---- FP32 WMMA GEMM: 16 rows x 128 classes per wave ----
    v8f acc[8];
    #pragma unroll
    for (int t = 0; t < 8; ++t)
        #pragma unroll
        for (int i = 0; i < 8; ++i) acc[t][i] = 0.0f;

    const int    cbase = wave * 128;
    const float* mw    = m + (size_t)cbase * NDIM;

    for (int k0 = 0; k0 < NDIM; k0 += 4) {
        const int koff = k0 + half * 2;
        // A frag 16x4: lanes 0-15 hold K0,K1 ; lanes 16-31 hold K2,K3 (ISA layout)
        v2f a = *(const v2f*)(xtile + l16 * NDIM + koff);
        #pragma unroll
        for (int t = 0; t < 8; ++t) {
            // B frag 4x16: column = class (cbase+16t+l16), same K split as A
            v2f b = *(const v2f*)(mw + (size_t)(t * 16 + l16) * NDIM + koff);
            acc[t] = __builtin_amdgcn_wmma_f32_16x16x4_f32(
                false, a, false, b, (short)0, acc[t], false, false);
        }
    }

    // ---- logits = -sqrt(max(x2 - 2*dot + m2, 0)) / s  ->  padded LDS tile ----
    float x2v[8];
    #pragma unroll
    for (int v = 0; v < 8; ++v) x2v[v] = x2row[v + half * 8];

    #pragma unroll
    for (int t = 0; t < 8; ++t) {
        const int   c   = cbase + t * 16 + l16;
        const float m2c = m2[c];
        const float isc = invs[c];
        #pragma unroll
        for (int v = 0; v < 8; ++v) {
            float d2 = x2v[v] - 2.0f * acc[t][v] + m2c;
            d2 = fmaxf(d2, 0.0f);
            lg[(v + half * 8) * LG_STRIDE + c] = -sqrtf(d2) * isc;
        }
    }
    __syncthreads();

    // ---- per-row softmax entropy: 16 threads per row, two passes ----
    const int    r   = tid >> 4;
    const int    cl  = tid & 15;
    const float* row = lg + r * LG_STRIDE;

    float mx = -3.0e38f;
    for (int j = cl; j < NCLASS; j += 16) mx = fmaxf(mx, row[j]);
    float* redM = xtile;                  // x tile is dead now; reuse
    redM[r * 16 + cl] = mx;
    __syncthreads();
    float rmax = redM[r * 16];
    #pragma unroll
    for (int i = 1; i < 16; ++i) rmax = fmaxf(rmax, redM[r * 16 + i]);

    float Z = 0.0f, S = 0.0f;
    for (int j = cl; j < NCLASS; j += 16) {
        float v = row[j];
        float e = __expf(v - rmax);
        Z += e;
        S += e * v;
    }
    float* redZ = xtile + 256;
    float* redS = xtile + 512;
    float* redH = xtile + 768;
    redZ[r * 16 + cl] = Z;
    redS[r * 16 + cl] = S;
    __syncthreads();

    if (cl == 0) {
        float tZ = 0.0f, tS = 0.0f;
        #pragma unroll
        for (int i = 0; i < 16; ++i) { tZ += redZ[r * 16 + i]; tS += redS[r * 16 + i]; }
        // H = a + log Z - (1/Z) * sum e^(l-a) * l
        redH[r] = rmax + __logf(tZ) - tS / tZ;
    }
    __syncthreads();

    if (tid == 0) {
        float hs = 0.0f;
        #pragma unroll
        for (int i = 0; i < 16; ++i) hs += redH[i];
        partial[blockIdx.x] = hs;         // one coalesced store per block
    }
}

// ---------------------------------------------------------------------------
// Final: out = -(mean entropy + inter). Deterministic tree reduction.
// ---------------------------------------------------------------------------
__global__ void finalize_kernel(const float* __restrict__ partial,
                                const float* __restrict__ interv,
                                float* __restrict__ out) {
    __shared__ float red[256];
    float s = 0.0f;
    for (int i = threadIdx.x; i < NBLOCKS; i += 256) s += partial[i];
    red[threadIdx.x] = s;
    __syncthreads();
    #pragma unroll
    for (int off = 128; off > 0; off >>= 1) {
        if ((int)threadIdx.x < off) red[threadIdx.x] += red[threadIdx.x + off];
        __syncthreads();
    }
    if (threadIdx.x == 0)
        out[0] = -(red[0] * (1.0f / (float)NROWS) + interv[0]);
}

// ---------------------------------------------------------------------------
extern "C" void kernel_launch(void* const* d_in, const int* in_sizes, int n_in,
                              void* d_out, int out_size, void* d_ws, size_t ws_size,
                              hipStream_t stream) {
    const float* x = (const float*)d_in[0];   // [N, D]
    const float* m = (const float*)d_in[1];   // [K, D]
    const float* s = (const float*)d_in[2];   // [K]
    float* out = (float*)d_out;

    float* m2      = (float*)d_ws;            // [1024]
    float* invs    = m2 + NCLASS;             // [1024]
    float* interv  = invs + NCLASS;           // [1]
    float* partial = interv + 16;             // [8192] (16-elt aligned)

    precompute_kernel<<<(NCLASS + 255) / 256, 256, 0, stream>>>(m, s, m2, invs);
    inter_kernel<<<1, NDIM, 0, stream>>>(m, interv);

    const size_t smem_bytes = (size_t)(2048 + 16 + 16 * LG_STRIDE) * sizeof(float);
    entropy_wmma_kernel<<<NBLOCKS, 256, smem_bytes, stream>>>(x, m, m2, invs, partial);

    finalize_kernel<<<1, 256, 0, stream>>>(partial, interv, out);
}